// Sims_4810363372727
// MI455X (gfx1250) — compile-verified
//
#include <hip/hip_runtime.h>

typedef __attribute__((ext_vector_type(2))) float v2f;
typedef __attribute__((ext_vector_type(8))) float v8f;
typedef __attribute__((ext_vector_type(4))) int   v4i;

#define N_B    8
#define S_DIM  2048
#define E_DIM  512
#define TEMP_INV (1.0f / 13.544f)

#define M_TILE 32          // rows per workgroup
#define NBLK   128         // column block (8 waves x 16-col tiles)
#define KC     64          // K chunk staged in LDS (double buffered)
#define NKC    (E_DIM / KC)            // 8 chunks
#define AS_STRIDE 516      // 512 + 4 pad -> stride mod 64 == 4 (conflict-free)
#define BS_STRIDE 68       // 64 + 4 pad  -> stride mod 64 == 4

// ---------------------------------------------------------------------------
// CDNA5 async global->LDS copy (ASYNCcnt-tracked, no VGPR data movement).
// Builtin wants typed int4 pointers: (v4i AS1*, v4i AS3*, imm off, imm cpol).
// ---------------------------------------------------------------------------
typedef __attribute__((address_space(1))) v4i glb_v4i;
typedef __attribute__((address_space(3))) v4i lds_v4i;

__device__ __forceinline__ void async_copy_b128(const float* gsrc, float* ldst) {
    __builtin_amdgcn_global_load_async_to_lds_b128(
        (glb_v4i*)gsrc, (lds_v4i*)ldst, 0, 0);
}
__device__ __forceinline__ void wait_async0() {
    asm volatile("s_wait_asynccnt 0x0" ::: "memory");
}

// ---------------------------------------------------------------------------
// Kernel 1: per-row squared L2 norms. One wave (32 lanes) per row of 512 f32.
// ---------------------------------------------------------------------------
__global__ void __launch_bounds__(256) norms_kernel(const float* __restrict__ x,
                                                    float* __restrict__ norms) {
    const int lane = threadIdx.x & 31;
    const int wave = threadIdx.x >> 5;
    const int row  = blockIdx.x * 8 + wave;          // 0 .. N_B*S_DIM-1
    const float4* x4 = (const float4*)x + (size_t)row * (E_DIM / 4);
    float s = 0.f;
#pragma unroll
    for (int j = 0; j < 4; ++j) {
        float4 v = x4[lane + 32 * j];
        s += v.x * v.x + v.y * v.y + v.z * v.z + v.w * v.w;
    }
#pragma unroll
    for (int off = 16; off > 0; off >>= 1) s += __shfl_xor(s, off, 32);
    if (lane == 0) norms[row] = s;
}

// ---------------------------------------------------------------------------
// Kernel 2: fused Gram (WMMA f32 16x16x4) + exp(-dist/T) + row-sum partials.
// B chunks are staged with double-buffered async global->LDS copies.
// ---------------------------------------------------------------------------
__global__ void __launch_bounds__(256) gram_softmax_kernel(
        const float* __restrict__ x, const float* __restrict__ norms,
        float* __restrict__ rowsums, float* __restrict__ out) {
    __shared__ float As[M_TILE * AS_STRIDE];       // full 32 x 512 A slab
    __shared__ float Bs[2][NBLK * BS_STRIDE];      // double-buffered 128 x 64
    __shared__ float sqRow[M_TILE];
    __shared__ float rowSum[M_TILE];

    const int b       = blockIdx.y;
    const int rowBase = blockIdx.x * M_TILE;
    const int t       = threadIdx.x;
    const int lane    = t & 31;
    const int wave    = t >> 5;                    // column tile within block
    const int half    = lane >> 4;                 // 0: K+0..1, 1: K+2..3
    const int l15     = lane & 15;

    const float* xb = x + (size_t)b * S_DIM * E_DIM;
    const float* nb = norms + (size_t)b * S_DIM;

    // ---- stage A slab (32 x 512) once; init sqRow / rowSum ----
    {
        const float4* g = (const float4*)(xb + (size_t)rowBase * E_DIM);
#pragma unroll
        for (int i = 0; i < (M_TILE * E_DIM / 4) / 256; ++i) {   // 16 iters
            int fid = i * 256 + t;
            int r   = fid >> 7;          // 128 float4 per row
            int c4  = fid & 127;
            float4 v = g[r * 128 + c4];
            *(float4*)&As[r * AS_STRIDE + c4 * 4] = v;
        }
        if (t < M_TILE)                 sqRow[t]       = nb[rowBase + t];
        if (t >= 64 && t < 64 + M_TILE) rowSum[t - 64] = 0.f;
    }
    __syncthreads();

    v8f  acc[2];
    float rsum[16];
#pragma unroll
    for (int i = 0; i < 16; ++i) rsum[i] = 0.f;
    const v8f vzero = {};

    for (int jb = 0; jb < S_DIM / NBLK; ++jb) {
        const int j0 = jb * NBLK;
        acc[0] = vzero;
        acc[1] = vzero;

        // issue chunk 0 into buffer 0 (buffer 0 unread since a barrier ago)
#pragma unroll
        for (int i = 0; i < 8; ++i) {
            int fid = i * 256 + t;
            int r   = fid >> 4;        // 16 float4 per row
            int c4  = fid & 15;
            async_copy_b128(xb + (size_t)(j0 + r) * E_DIM + 0 * KC + c4 * 4,
                            &Bs[0][r * BS_STRIDE + c4 * 4]);
        }

        for (int kc = 0; kc < NKC; ++kc) {
            wait_async0();             // this wave's chunk-kc copies done
            __syncthreads();           // everyone's copies done / prev compute done

            if (kc + 1 < NKC) {        // prefetch next chunk into idle buffer
                const int kb2 = (kc + 1) * KC;
#pragma unroll
                for (int i = 0; i < 8; ++i) {
                    int fid = i * 256 + t;
                    int r   = fid >> 4;
                    int c4  = fid & 15;
                    async_copy_b128(xb + (size_t)(j0 + r) * E_DIM + kb2 + c4 * 4,
                                    &Bs[(kc + 1) & 1][r * BS_STRIDE + c4 * 4]);
                }
            }

            const float* aBase = &As[l15 * AS_STRIDE + kc * KC + half * 2];
            const float* bBase = &Bs[kc & 1][(wave * 16 + l15) * BS_STRIDE + half * 2];
#pragma unroll 4
            for (int k4 = 0; k4 < KC / 4; ++k4) {
                // B frag (4x16): lane n holds B[2h][n], B[2h+1][n]
                v2f bf;
                bf.x = bBase[k4 * 4];
                bf.y = bBase[k4 * 4 + 1];
#pragma unroll
                for (int rt = 0; rt < 2; ++rt) {
                    // A frag (16x4): lane m holds A[m][2h], A[m][2h+1]
                    v2f af;
                    af.x = aBase[rt * 16 * AS_STRIDE + k4 * 4];
                    af.y = aBase[rt * 16 * AS_STRIDE + k4 * 4 + 1];
                    acc[rt] = __builtin_amdgcn_wmma_f32_16x16x4_f32(
                        false, af, false, bf, (short)0, acc[rt], false, false);
                }
            }
        }

        // ---- epilogue: dist -> exp(-dist/T), store E, row-sum partials ----
        const int   gcol = j0 + wave * 16 + l15;
        const float sb   = nb[gcol];
#pragma unroll
        for (int rt = 0; rt < 2; ++rt) {
#pragma unroll
            for (int v = 0; v < 8; ++v) {
                const int   mloc = rt * 16 + half * 8 + v;   // C layout: M = v + 8*half
                const float g2   = acc[rt][v];
                const float sa   = sqRow[mloc];
                // -dist/T = (2g - ||a||^2 - ||b||^2)/T, always <= ~0
                const float e = __expf((2.0f * g2 - sa - sb) * TEMP_INV);
                out[((size_t)b * S_DIM + rowBase + mloc) * S_DIM + gcol] = e;
                rsum[rt * 8 + v] += e;
            }
        }
    }

    // ---- fold per-lane partials: reduce across the 16 lanes of each half ----
#pragma unroll
    for (int i = 0; i < 16; ++i) {
        float r = rsum[i];
        r += __shfl_xor(r, 1, 32);
        r += __shfl_xor(r, 2, 32);
        r += __shfl_xor(r, 4, 32);
        r += __shfl_xor(r, 8, 32);
        if (l15 == 0) {
            const int rt = i >> 3, v = i & 7;
            atomicAdd(&rowSum[rt * 16 + half * 8 + v], r);
        }
    }
    __syncthreads();
    if (t < M_TILE) rowsums[(size_t)b * S_DIM + rowBase + t] = rowSum[t];
}

// ---------------------------------------------------------------------------
// Kernel 3: out *= 1/rowsum (float4 per thread; 512 float4 per row).
// ---------------------------------------------------------------------------
__global__ void __launch_bounds__(256) scale_kernel(float* __restrict__ out,
                                                    const float* __restrict__ rowsums) {
    const size_t i   = (size_t)blockIdx.x * blockDim.x + threadIdx.x;  // float4 idx
    const int    row = (int)(i >> 9);                                  // /512
    const float  inv = 1.0f / rowsums[row];
    float4* o4 = (float4*)out;
    float4  v  = o4[i];
    v.x *= inv; v.y *= inv; v.z *= inv; v.w *= inv;
    o4[i] = v;
}

// ---------------------------------------------------------------------------
extern "C" void kernel_launch(void* const* d_in, const int* in_sizes, int n_in,
                              void* d_out, int out_size, void* d_ws, size_t ws_size,
                              hipStream_t stream) {
    (void)in_sizes; (void)n_in; (void)out_size; (void)ws_size;
    const float* x   = (const float*)d_in[0];
    float* out       = (float*)d_out;
    float* norms     = (float*)d_ws;                 // N_B*S_DIM floats
    float* rowsums   = norms + N_B * S_DIM;          // N_B*S_DIM floats

    norms_kernel<<<dim3(N_B * S_DIM / 8), 256, 0, stream>>>(x, norms);
    gram_softmax_kernel<<<dim3(S_DIM / M_TILE, N_B), 256, 0, stream>>>(
        x, norms, rowsums, out);
    const unsigned nScaleBlocks =
        (unsigned)(((size_t)N_B * S_DIM * S_DIM / 4) / 256);
    scale_kernel<<<dim3(nScaleBlocks), 256, 0, stream>>>(out, rowsums);
}